// gap_2370821948148
// MI455X (gfx1250) — compile-verified
//
#include <hip/hip_runtime.h>
#include <math.h>

#define N_BATCH 4096
#define KP 64
#define NP 65              // KP + 1
#define GAMMA_C 0.5f
#define POS_R2 0.36f       // 0.6^2
#define NEG_R2 1.44f       // 1.2^2
#define BIG_C  1000000000000.0f

typedef __attribute__((ext_vector_type(2))) float v2f;
typedef __attribute__((ext_vector_type(8))) float v8f;

#if __has_builtin(__builtin_amdgcn_global_load_async_to_lds_b128) && \
    __has_builtin(__builtin_amdgcn_global_load_async_to_lds_b32) &&  \
    __has_builtin(__builtin_amdgcn_s_wait_asynccnt)
#define USE_ASYNC_LDS 1
// Builtin signature (from hipcc diagnostic):
//   b128: (int __vector(4) __device__*, int __vector(4) __shared__*, Ii, Ii)
//   b32 : (int __device__*,             int __shared__*,             Ii, Ii)
typedef int v4i_t __attribute__((vector_size(16)));
typedef __attribute__((address_space(1))) v4i_t* as1_v4i;
typedef __attribute__((address_space(3))) v4i_t* as3_v4i;
typedef __attribute__((address_space(1))) int*   as1_i;
typedef __attribute__((address_space(3))) int*   as3_i;
#endif

__device__ __forceinline__ float wred_f(float v) {
#pragma unroll
  for (int m = 16; m >= 1; m >>= 1) v += __shfl_xor(v, m, 32);
  return v;
}
__device__ __forceinline__ int wred_i(int v) {
#pragma unroll
  for (int m = 16; m >= 1; m >>= 1) v += __shfl_xor(v, m, 32);
  return v;
}

// One workgroup per batch. 512 threads = 16 wave32 waves.
__global__ __launch_bounds__(512) void circle_loss_main(
    const float* __restrict__ pos_pts,   // [B][64][3]
    const float* __restrict__ anc_pts,   // [B][64][3]
    const int*   __restrict__ pos_mask,  // [B][64]  (bool -> int)
    const int*   __restrict__ anc_mask,  // [B][64]
    const float* __restrict__ ms,        // [B][65][65]
    const float* __restrict__ xf,        // [4][4] row-major
    float* __restrict__ partial)         // [B][4]
{
  __shared__ float s_ms[NP * NP];          // 16900 B, row stride 65 dwords
  __shared__ float s_pos[KP][4];           // xyz + 0-pad (K=4 for WMMA)
  __shared__ float s_anc[KP][4];
  __shared__ float s_x2[KP];
  __shared__ float s_y2[KP];
  __shared__ int   s_pm[KP];
  __shared__ int   s_am[KP];
  __shared__ unsigned char s_flags[KP][68]; // bit0=corr, bit1=neg; 68-byte rows
  __shared__ float s_red[16][4];

  const int b    = blockIdx.x;
  const int tid  = threadIdx.x;
  const int lane = tid & 31;
  const int wave = tid >> 5;

  const float* msb = ms + (size_t)b * (NP * NP);

  // ---- Stage matching_scores: async global->LDS DMA (no VGPR roundtrip) ----
#ifdef USE_ASYNC_LDS
  {
    const v4i_t* g4 = (const v4i_t*)msb;
    v4i_t* l4 = (v4i_t*)s_ms;
    // 4225 floats = 16900 B = 1056 x 16 B + 4 B tail
    for (int idx = tid; idx < 1056; idx += 512) {
      __builtin_amdgcn_global_load_async_to_lds_b128(
          (as1_v4i)(g4 + idx), (as3_v4i)(l4 + idx), 0, 0);
    }
    if (tid == 0) {
      __builtin_amdgcn_global_load_async_to_lds_b32(
          (as1_i)(msb + 4224), (as3_i)(s_ms + 4224), 0, 0);
    }
  }
#else
  for (int idx = tid; idx < NP * NP; idx += 512) s_ms[idx] = msb[idx];
#endif

  // ---- Stage points (apply transform to anchors), masks ----
  if (tid < KP) {
    const int i = tid;
    const float* pp = pos_pts + ((size_t)b * KP + i) * 3;
    float px = pp[0], py = pp[1], pz = pp[2];
    s_pos[i][0] = px; s_pos[i][1] = py; s_pos[i][2] = pz; s_pos[i][3] = 0.0f;
    s_x2[i] = px * px + py * py + pz * pz;

    const float* ap = anc_pts + ((size_t)b * KP + i) * 3;
    float ax = ap[0], ay = ap[1], az = ap[2];
    // out[e] = dot(R[e,:], p) + t[e],  R = xf[:3,:3], t = xf[:3,3]
    float tx = xf[0] * ax + xf[1] * ay + xf[2]  * az + xf[3];
    float ty = xf[4] * ax + xf[5] * ay + xf[6]  * az + xf[7];
    float tz = xf[8] * ax + xf[9] * ay + xf[10] * az + xf[11];
    s_anc[i][0] = tx; s_anc[i][1] = ty; s_anc[i][2] = tz; s_anc[i][3] = 0.0f;
    s_y2[i] = tx * tx + ty * ty + tz * tz;

    s_pm[i] = pos_mask[(size_t)b * KP + i];
    s_am[i] = anc_mask[(size_t)b * KP + i];
  }

#ifdef USE_ASYNC_LDS
  __builtin_amdgcn_s_wait_asynccnt(0);   // own async loads complete
#endif
  __syncthreads();                       // all waves' staging visible

  // ---- Gram matrix via V_WMMA_F32_16X16X4_F32: one 16x16 tile per wave ----
  {
    const int ti = wave >> 2;          // row tile
    const int tj = wave & 3;           // col tile
    const int mrow = ti * 16 + (lane & 15);
    const int ncol = tj * 16 + (lane & 15);
    const int k0 = (lane < 16) ? 0 : 2;           // A/B K split across lane halves

    v2f a, bf;
    a.x  = s_pos[mrow][k0];  a.y  = s_pos[mrow][k0 + 1];   // A: 16x4 (K pad=0)
    bf.x = s_anc[ncol][k0];  bf.y = s_anc[ncol][k0 + 1];   // B: 4x16

    v8f c = {};
    c = __builtin_amdgcn_wmma_f32_16x16x4_f32(
        /*neg_a=*/false, a, /*neg_b=*/false, bf,
        /*c_mod=*/(short)0, c, /*reuse_a=*/false, /*reuse_b=*/false);

    // C layout: VGPR v -> M = v (lanes 0..15) or v+8 (lanes 16..31), N = lane%16
    const int jj   = tj * 16 + (lane & 15);
    const int ibas = ti * 16 + ((lane >= 16) ? 8 : 0);
    const int amj_nz = (s_am[jj] != 0) ? 1 : 0;
    const float y2 = s_y2[jj];
#pragma unroll
    for (int v = 0; v < 8; ++v) {
      const int ii = ibas + v;
      const float d = s_x2[ii] - 2.0f * c[v] + y2;
      const int pm_nz = (s_pm[ii] != 0) ? 1 : 0;       // unconditional LDS read
      const int corr  = ((d < POS_R2) ? 1 : 0) & pm_nz & amj_nz; // branch-free
      const int neg   = ((d > NEG_R2) ? 1 : 0) << 1;
      s_flags[ii][jj] = (unsigned char)(corr | neg);
    }
  }
  __syncthreads();

  float acc1 = 0.0f, k1 = 0.0f, acc2 = 0.0f, k2 = 0.0f;

  // ---- Loss1: rows i in 0..63 over 65 columns; wave handles 4 rows ----
#pragma unroll
  for (int rr = 0; rr < 4; ++rr) {
    const int i = wave + rr * 16;
    const unsigned char* fr = s_flags[i];
    const float* mr = &s_ms[i * NP];
    const int f1 = fr[lane], f2 = fr[lane + 32];
    const float m1 = mr[lane], m2 = mr[lane + 32];

    int   cc = (f1 & 1) + (f2 & 1);
    float cs = ((f1 & 1) ? -m1 : 0.0f) + ((f2 & 1) ? -m2 : 0.0f);
    cc = wred_i(cc);
    cs = wred_f(cs);
    const bool row_none = (cc == 0);
    const float mlast = mr[KP];                       // ms[i][64]
    const float pcnt = (float)cc + (row_none ? 1.0f : 0.0f);
    const float psum = cs + (row_none ? -mlast : 0.0f);
    const float pscore = psum / pcnt;                 // pcnt >= 1 always

    float ns1 = (f1 & 2) ? -m1 : BIG_C;
    float ns2 = (f2 & 2) ? -m2 : BIG_C;
    float t = fmaxf(pscore - ns1 + GAMMA_C, 0.0f)
            + fmaxf(pscore - ns2 + GAMMA_C, 0.0f);
    t = wred_f(t);
    const float nslast = (!row_none) ? -mlast : BIG_C;
    t += fmaxf(pscore - nslast + GAMMA_C, 0.0f);

    const float per = logf(t + 1.0f);
    if (pscore != BIG_C) { acc1 += per; k1 += 1.0f; }
  }

  // ---- Loss2: columns j in 0..63 over 65 rows; wave handles 4 columns ----
#pragma unroll
  for (int rr = 0; rr < 4; ++rr) {
    const int j = wave + rr * 16;
    const int f1 = s_flags[lane][j], f2 = s_flags[lane + 32][j];
    const float m1 = s_ms[lane * NP + j], m2 = s_ms[(lane + 32) * NP + j];

    int   cc = (f1 & 1) + (f2 & 1);
    float cs = ((f1 & 1) ? -m1 : 0.0f) + ((f2 & 1) ? -m2 : 0.0f);
    cc = wred_i(cc);
    cs = wred_f(cs);
    const bool col_none = (cc == 0);
    const float mlast = s_ms[KP * NP + j];            // ms[64][j]
    const float pcnt = (float)cc + (col_none ? 1.0f : 0.0f);
    const float psum = cs + (col_none ? -mlast : 0.0f);
    const float pscore = psum / pcnt;

    float ns1 = (f1 & 2) ? -m1 : BIG_C;
    float ns2 = (f2 & 2) ? -m2 : BIG_C;
    float t = fmaxf(pscore - ns1 + GAMMA_C, 0.0f)
            + fmaxf(pscore - ns2 + GAMMA_C, 0.0f);
    t = wred_f(t);
    const float nslast = (!col_none) ? -mlast : BIG_C;
    t += fmaxf(pscore - nslast + GAMMA_C, 0.0f);

    const float per = logf(t + 1.0f);
    if (pscore != BIG_C) { acc2 += per; k2 += 1.0f; }
  }

  if (lane == 0) {
    s_red[wave][0] = acc1; s_red[wave][1] = k1;
    s_red[wave][2] = acc2; s_red[wave][3] = k2;
  }
  __syncthreads();
  if (tid < 4) {
    float s = 0.0f;
#pragma unroll
    for (int w = 0; w < 16; ++w) s += s_red[w][tid];
    partial[(size_t)b * 4 + tid] = s;
  }
}

// Deterministic final reduction of [B][4] partials -> scalar loss
__global__ __launch_bounds__(256) void circle_loss_finalize(
    const float* __restrict__ partial, float* __restrict__ out)
{
  __shared__ float s_f[8][4];
  const int tid = threadIdx.x;
  const int lane = tid & 31;
  const int wave = tid >> 5;
  float a1 = 0.0f, c1 = 0.0f, a2 = 0.0f, c2 = 0.0f;
  for (int i = tid; i < N_BATCH; i += 256) {
    a1 += partial[(size_t)i * 4 + 0];
    c1 += partial[(size_t)i * 4 + 1];
    a2 += partial[(size_t)i * 4 + 2];
    c2 += partial[(size_t)i * 4 + 3];
  }
  a1 = wred_f(a1); c1 = wred_f(c1); a2 = wred_f(a2); c2 = wred_f(c2);
  if (lane == 0) { s_f[wave][0] = a1; s_f[wave][1] = c1; s_f[wave][2] = a2; s_f[wave][3] = c2; }
  __syncthreads();
  if (tid == 0) {
    float A1 = 0, C1 = 0, A2 = 0, C2 = 0;
#pragma unroll
    for (int w = 0; w < 8; ++w) { A1 += s_f[w][0]; C1 += s_f[w][1]; A2 += s_f[w][2]; C2 += s_f[w][3]; }
    const float loss1 = A1 / fmaxf(C1, 1.0f);
    const float loss2 = A2 / fmaxf(C2, 1.0f);
    out[0] = 0.5f * (loss1 + loss2);
  }
}

extern "C" void kernel_launch(void* const* d_in, const int* in_sizes, int n_in,
                              void* d_out, int out_size, void* d_ws, size_t ws_size,
                              hipStream_t stream) {
  (void)in_sizes; (void)n_in; (void)out_size; (void)ws_size;
  const float* pos_pts  = (const float*)d_in[0];
  const float* anc_pts  = (const float*)d_in[1];
  const int*   pos_mask = (const int*)d_in[2];
  const int*   anc_mask = (const int*)d_in[3];
  const float* mscores  = (const float*)d_in[4];
  const float* xform    = (const float*)d_in[5];
  float* partial = (float*)d_ws;                   // 4096*4 floats = 64 KB
  float* out = (float*)d_out;

  circle_loss_main<<<N_BATCH, 512, 0, stream>>>(
      pos_pts, anc_pts, pos_mask, anc_mask, mscores, xform, partial);
  circle_loss_finalize<<<1, 256, 0, stream>>>(partial, out);
}